// CPDP_AM_net_7009386627331
// MI455X (gfx1250) — compile-verified
//
#include <hip/hip_runtime.h>
#include <hip/hip_bf16.h>

// ---------------------------------------------------------------------------
// CDNA5 / gfx1250 attention-decoder kernel (wave32).
//   grid = 256 workgroups (one per batch element), block = 256 threads = 8 waves.
//   Memory-bound: streams K_att/V_att (1.57 GB) once via b128/b64 coalesced rows.
//   Projections (W0 / Wqf, 512x512 matvec) run on V_WMMA_F32_16X16X4_F32
//   with the input vector replicated across the 16 N columns (exact f32).
// ---------------------------------------------------------------------------

#define DIM       512
#define NN        500
#define NHEADS    16
#define KVSTRIDE  1536      // DIM * NB_LAYERS
#define SCN       516       // padded score row stride (bank-conflict-free head stride)
#define NEG_INF   (-1000000000.0f)
#define RSQRT32   0.17677669529663687f   // 1/sqrt(32)
#define RSQRT512  0.044194173824159216f  // 1/sqrt(512)

typedef __attribute__((ext_vector_type(2))) float v2f;
typedef __attribute__((ext_vector_type(8))) float v8f;

__device__ __forceinline__ float wave_sum(float v) {
#pragma unroll
    for (int o = 16; o > 0; o >>= 1) v += __shfl_xor(v, o, 32);
    return v;
}
__device__ __forceinline__ float wave_max(float v) {
#pragma unroll
    for (int o = 16; o > 0; o >>= 1) v = fmaxf(v, __shfl_xor(v, o, 32));
    return v;
}

// y[0..511] = W(512x512, row-major) * x[0..511] + bias, via f32 WMMA.
// B is the x-vector broadcast across all 16 N columns; column 0 of D is exact.
// A layout per ISA (32-bit A 16x4): lanes 0-15 hold K={0,1}, lanes 16-31 K={2,3}.
__device__ __forceinline__ void proj512_wmma(const float* __restrict__ W,
                                             const float* __restrict__ bias,
                                             const float* x,   // LDS
                                             float* y,         // LDS
                                             int wid, int lane)
{
    const int m  = lane & 15;
    const int hi = lane >> 4;            // 0: K pair {0,1}; 1: K pair {2,3}
    for (int tile = wid; tile < 32; tile += 8) {   // uniform across waves: EXEC all-1s
        const int i0 = tile * 16;
        const float* wr = W + (size_t)(i0 + m) * DIM;
        v8f c = {0.f, 0.f, 0.f, 0.f, 0.f, 0.f, 0.f, 0.f};
#pragma unroll 8
        for (int k0 = 0; k0 < DIM; k0 += 4) {
            const int ka = k0 + hi * 2;
            v2f a; a.x = wr[ka]; a.y = wr[ka + 1];
            v2f bb; bb.x = x[ka]; bb.y = x[ka + 1];   // replicated across N columns
            c = __builtin_amdgcn_wmma_f32_16x16x4_f32(
                    false, a, false, bb, (short)0, c, false, false);
        }
        // D layout: VGPR r -> M=r (lanes 0-15) / M=8+r (lanes 16-31), N=lane%16.
        if (m == 0) {
#pragma unroll
            for (int r = 0; r < 8; ++r) {
                const int i = i0 + hi * 8 + r;
                y[i] = c[r] + bias[i];
            }
        }
    }
}

__global__ __launch_bounds__(256)
void attn_decoder_kernel(const float* __restrict__ query,
                         const float* __restrict__ K_att,
                         const float* __restrict__ V_att,
                         const unsigned char* __restrict__ mask,
                         const float* __restrict__ W0_w,
                         const float* __restrict__ W0_b,
                         const float* __restrict__ Wqf_w,
                         const float* __restrict__ Wqf_b,
                         float* __restrict__ out)
{
    __shared__ float qv[DIM];            // current query vector
    __shared__ float ov[DIM];            // MHA output / projection input
    __shared__ float sc[NHEADS * SCN];   // scores / softmax weights

    const int b    = blockIdx.x;
    const int tid  = threadIdx.x;
    const int lane = tid & 31;
    const int wid  = tid >> 5;
    const size_t kvbase = (size_t)b * NN * KVSTRIDE;

    qv[tid]       = query[(size_t)b * DIM + tid];
    qv[tid + 256] = query[(size_t)b * DIM + tid + 256];
    __syncthreads();

    for (int l = 0; l < 3; ++l) {
        if (l == 2) {
            // q_final = Wqf * q + b  (copy q -> ov, project ov -> qv)
            ov[tid] = qv[tid]; ov[tid + 256] = qv[tid + 256];
            __syncthreads();
            proj512_wmma(Wqf_w, Wqf_b, ov, qv, wid, lane);
            __syncthreads();
        }

        const float* Kb = K_att + kvbase + l * DIM;

        // q chunk for this lane (loop-invariant over n): d = lane*16 + 0..15
        float4 qreg[4];
#pragma unroll
        for (int i = 0; i < 4; ++i)
            qreg[i] = ((const float4*)qv)[lane * 4 + i];

        if (l < 2) {
            // ---- 16-head scores: wave per node, 4x b128 per lane (2KB row) ----
            // lane covers d = lane*16..lane*16+15, entirely inside head lane/2.
            for (int n = wid; n < NN; n += 8) {
                const float* kr = Kb + (size_t)n * KVSTRIDE;
                if (n + 8 < NN) __builtin_prefetch(kr + 8 * KVSTRIDE, 0, 1);
                const bool msk = mask[b * NN + n] != 0;
                float p = 0.f;
#pragma unroll
                for (int i = 0; i < 4; ++i) {
                    const float4 kv4 = ((const float4*)kr)[lane * 4 + i];
                    p = fmaf(kv4.x, qreg[i].x, p);
                    p = fmaf(kv4.y, qreg[i].y, p);
                    p = fmaf(kv4.z, qreg[i].z, p);
                    p = fmaf(kv4.w, qreg[i].w, p);
                }
                p += __shfl_xor(p, 1, 32);       // combine lane pair -> head sum
                float s = p * RSQRT32;
                if (msk) s = NEG_INF;
                if ((lane & 1) == 0) sc[(lane >> 1) * SCN + n] = s;
            }
            __syncthreads();

            // ---- per-head softmax over 500 nodes (2 heads per wave) ----
            for (int h = wid; h < NHEADS; h += 8) {
                float m = -3.0e38f;
                for (int n = lane; n < NN; n += 32) m = fmaxf(m, sc[h * SCN + n]);
                m = wave_max(m);
                float ssum = 0.f;
                for (int n = lane; n < NN; n += 32) {
                    const float e = __expf(sc[h * SCN + n] - m);
                    sc[h * SCN + n] = e;
                    ssum += e;
                }
                ssum = wave_sum(ssum);
                const float inv = 1.0f / ssum;
                for (int n = lane; n < NN; n += 32) sc[h * SCN + n] *= inv;
            }
            __syncthreads();

            // ---- out[d] = sum_n w[head(d)][n] * V[n][d]; b64 per thread ----
            const float* Vb = V_att + kvbase + l * DIM;
            const int d = tid * 2;             // pair stays inside one head
            const int hd = d >> 5;
            float a0 = 0.f, a1 = 0.f;
            for (int n = 0; n < NN; ++n) {
                const float* vr = Vb + (size_t)n * KVSTRIDE;
                if (n + 1 < NN) __builtin_prefetch(vr + KVSTRIDE, 0, 1);
                const float2 v2 = *(const float2*)(vr + d);
                const float w = sc[hd * SCN + n];
                a0 = fmaf(w, v2.x, a0);
                a1 = fmaf(w, v2.y, a1);
            }
            ov[d] = a0; ov[d + 1] = a1;
            __syncthreads();

            // ---- query = W0 * out + b0 (WMMA) ----
            proj512_wmma(W0_w, W0_b, ov, qv, wid, lane);
            __syncthreads();
        } else {
            // ---- final layer: single head (dh=512), 10*tanh clip, mask ----
            for (int n = wid; n < NN; n += 8) {
                const float* kr = Kb + (size_t)n * KVSTRIDE;
                if (n + 8 < NN) __builtin_prefetch(kr + 8 * KVSTRIDE, 0, 1);
                float p = 0.f;
#pragma unroll
                for (int i = 0; i < 4; ++i) {
                    const float4 kv4 = ((const float4*)kr)[lane * 4 + i];
                    p = fmaf(kv4.x, qreg[i].x, p);
                    p = fmaf(kv4.y, qreg[i].y, p);
                    p = fmaf(kv4.z, qreg[i].z, p);
                    p = fmaf(kv4.w, qreg[i].w, p);
                }
                float s = wave_sum(p) * RSQRT512;
                s = 10.0f * tanhf(s);
                if (mask[b * NN + n]) s = NEG_INF;
                if (lane == 0) sc[n] = s;
            }
            __syncthreads();
            if (wid == 0) {
                float m = -3.0e38f;
                for (int n = lane; n < NN; n += 32) m = fmaxf(m, sc[n]);
                m = wave_max(m);
                float ssum = 0.f;
                for (int n = lane; n < NN; n += 32) {
                    const float e = __expf(sc[n] - m);
                    sc[n] = e;
                    ssum += e;
                }
                ssum = wave_sum(ssum);
                const float inv = 1.0f / ssum;
                for (int n = lane; n < NN; n += 32) sc[n] *= inv;
            }
            __syncthreads();
            for (int n = tid; n < NN; n += 256)
                out[(size_t)b * NN + n] = sc[n];
        }
    }
}

extern "C" void kernel_launch(void* const* d_in, const int* in_sizes, int n_in,
                              void* d_out, int out_size, void* d_ws, size_t ws_size,
                              hipStream_t stream)
{
    const float*         query = (const float*)d_in[0];
    const float*         K_att = (const float*)d_in[1];
    const float*         V_att = (const float*)d_in[2];
    const unsigned char* mask  = (const unsigned char*)d_in[3];
    const float*         W0_w  = (const float*)d_in[4];
    const float*         W0_b  = (const float*)d_in[5];
    const float*         Wqf_w = (const float*)d_in[6];
    const float*         Wqf_b = (const float*)d_in[7];

    attn_decoder_kernel<<<256, 256, 0, stream>>>(
        query, K_att, V_att, mask, W0_w, W0_b, Wqf_w, Wqf_b, (float*)d_out);
}